// CDF_quadratic_34522947125489
// MI455X (gfx1250) — compile-verified
//
#include <hip/hip_runtime.h>
#include <math.h>

#define N_BINS 64
#define DIM    8
#define RGEO   1.2f
#define BOUNDF 50.0f

typedef __attribute__((ext_vector_type(2))) float v2f;
typedef __attribute__((ext_vector_type(8))) float v8f;

// Workspace layout (floats):
//   ws[0] = C1 = (r-1)/x1L_n  (== 2*(r^32 - 1))
//   ws[1] = 1/log2(r)  (binning uses hardware log2)
//   ws[8 ..) : tab[d*64 + k] = float4{ v1, slope, F_pre, mesh_k }   (512 float4 = 8KB)

// ---------------------------------------------------------------------------
// Setup: build tables from p (63x8). Single wave (32 threads, EXEC all ones).
// Cumulative sum F = StrictLowerOnes(64x64) x cell(64x8) done with
// V_WMMA_F32_16X16X4_F32 (f32 WMMA -> exact precision).
// cell is staged padded to 16 columns so all 32 lanes load B unconditionally
// (no EXEC-predicated ds_loads around the WMMA region).
// ---------------------------------------------------------------------------
__global__ __launch_bounds__(32)
void cdfq_setup_kernel(const float* __restrict__ p, float* __restrict__ ws) {
  __shared__ float mesh_s[N_BINS + 1];
  __shared__ float elmt_s[N_BINS];
  __shared__ float pdf_s[(N_BINS + 1) * DIM];
  __shared__ float cell_p[N_BINS * 16];     // padded to 16 cols (8..15 = 0)
  __shared__ float F_s[N_BINS * DIM];

  const int lane = threadIdx.x;
  const float r   = RGEO;
  const float rm  = powf(r, 32.0f);                 // r^(n_bins/2)
  const float x1L = BOUNDF * (r - 1.0f) / (rm - 1.0f);

  // geometric mesh on [0,1]
  for (int i = lane; i <= N_BINS; i += 32) {
    float v;
    if (i == 0)            v = 0.0f;
    else if (i == N_BINS)  v = 1.0f;
    else {
      float id  = (float)(i - N_BINS / 2);
      float mag = (1.0f - powf(r, fabsf(id))) / (1.0f - r);
      float xr  = (id >= 0.0f) ? x1L * mag : -x1L * mag;
      v = (xr + BOUNDF) * (0.5f / BOUNDF);
    }
    mesh_s[i] = v;
  }
  __syncthreads();
  for (int i = lane; i < N_BINS; i += 32) elmt_s[i] = mesh_s[i + 1] - mesh_s[i];
  __syncthreads();

  // pdf normalization: one lane per dim (tiny)
  if (lane < DIM) {
    const int d = lane;
    float s = 0.0f;
    for (int i = 0; i < N_BINS - 1; ++i)
      s += expf(p[i * DIM + d]) * (elmt_s[i] + elmt_s[i + 1]) * 0.5f;
    const float scale = (1.0f - elmt_s[0]) / s;
    pdf_s[d] = 1.0f;
    for (int i = 0; i < N_BINS - 1; ++i)
      pdf_s[(i + 1) * DIM + d] = scale * expf(p[i * DIM + d]);
    pdf_s[N_BINS * DIM + d] = 1.0f;
  }
  __syncthreads();

  // cell[i][d] = (pdf[i]+pdf[i+1])*0.5*elmt[i], padded columns zeroed
  for (int idx = lane; idx < N_BINS * 16; idx += 32) {
    const int i = idx >> 4, d = idx & 15;
    cell_p[idx] = (d < DIM)
        ? (pdf_s[i * DIM + d] + pdf_s[(i + 1) * DIM + d]) * 0.5f * elmt_s[i]
        : 0.0f;
  }
  __syncthreads();

  // F = S x cell via f32 WMMA. A(16x4 f32): lanes0-15 carry K0/K1 in V0/V1,
  // lanes16-31 carry K2/K3. B(4x16) mirrors: N=lane&15, K = v + 2*(lane>>4).
  // C/D(16x16 f32): M = v + 8*(lane>>4), N = lane&15.
  {
    const int row = lane & 15;
    const int hi2 = (lane >> 4) * 2;
    const int n   = lane & 15;
    for (int tm = 0; tm < 4; ++tm) {
      v8f acc = {};
      const int rowM = tm * 16 + row;
      for (int tk = 0; tk < 16; ++tk) {
        const int k0 = tk * 4 + hi2;
        v2f a, b;
        a.x = (k0 + 0 < rowM) ? 1.0f : 0.0f;     // strictly-lower ones
        a.y = (k0 + 1 < rowM) ? 1.0f : 0.0f;
        b.x = cell_p[(k0 + 0) * 16 + n];         // unconditional, EXEC stays full
        b.y = cell_p[(k0 + 1) * 16 + n];
        acc = __builtin_amdgcn_wmma_f32_16x16x4_f32(false, a, false, b,
                                                    (short)0, acc, false, false);
      }
      if (n < DIM) {
        const int mbase = tm * 16 + (lane >> 4) * 8;
#pragma unroll
        for (int v = 0; v < 8; ++v)
          F_s[(mbase + v) * DIM + n] = acc[v];
      }
    }
  }
  __syncthreads();

  // Fused gather table, bank-friendly layout tab[d*64 + k]
  float4* tab_g = (float4*)(ws + 8);
  for (int idx = lane; idx < N_BINS * DIM; idx += 32) {
    const int k = idx & 63, d = idx >> 6;
    const float v1 = pdf_s[k * DIM + d];
    const float v2 = pdf_s[(k + 1) * DIM + d];
    float4 t;
    t.x = v1;
    t.y = (v2 - v1) / elmt_s[k];
    t.z = F_s[k * DIM + d];
    t.w = mesh_s[k];
    tab_g[idx] = t;
  }
  if (lane == 0) {
    ws[0] = 2.0f * (rm - 1.0f);                 // (r-1)/x1L_n
    ws[1] = 1.0f / log2f(r);                    // binning uses v_log_f32
  }
}

// ---------------------------------------------------------------------------
// Main pass: bandwidth-bound. 32B vector load / 32B+4B vector store per point,
// one ds_load_b128 gather per dim, one hw log2 per dim, one precise logf/point.
// ---------------------------------------------------------------------------
__global__ __launch_bounds__(256)
void cdfq_main_kernel(const float* __restrict__ x, const float* __restrict__ ws,
                      float* __restrict__ y, float* __restrict__ logdet, int npts) {
  __shared__ float4 tab[N_BINS * DIM];
  for (int idx = threadIdx.x; idx < N_BINS * DIM; idx += 256)
    tab[idx] = ((const float4*)(ws + 8))[idx];
  __syncthreads();

  const float C1       = ws[0];
  const float invlog2r = ws[1];
  const float invspan  = 1.0f / (2.0f * BOUNDF);

  int i = blockIdx.x * blockDim.x + threadIdx.x;
  const int stride = gridDim.x * blockDim.x;
  for (; i < npts; i += stride) {
    const float4* xin = (const float4*)(x + (size_t)i * DIM);
    const float4 xa = xin[0];
    const float4 xb = xin[1];
    float xv[8] = {xa.x, xa.y, xa.z, xa.w, xb.x, xb.y, xb.z, xb.w};
    float yv[8];
    float prod = 1.0f;
#pragma unroll
    for (int d = 0; d < 8; ++d) {
      const float xs = (xv[d] + BOUNDF) * invspan;
      const float t  = xs - 0.5f;
      // bin index on geometric mesh: m = floor(log_r(|t|*C1 + 1))
      const float a  = fabsf(t) * C1 + 1.0f;
      const int   mi = (int)floorf(__log2f(a) * invlog2r);
      const bool cover = (mi <= 31);
      int k = (t >= 0.0f) ? (32 + mi) : (31 - mi);
      k = min(max(k, 0), 63);
      const float4 tb = tab[d * 64 + k];          // {v1, slope, F_pre, mesh_k}
      const float xm   = xs - tb.w;
      float yy   = tb.z + xm * (xm * 0.5f * tb.y + tb.x);
      float dloc = xm * tb.y + tb.x;
      yy   = cover ? yy   : xs;
      dloc = cover ? dloc : 1.0f;
      prod *= dloc;
      yv[d] = yy * (2.0f * BOUNDF) - BOUNDF;
    }
    float4* yo = (float4*)(y + (size_t)i * DIM);
    yo[0] = make_float4(yv[0], yv[1], yv[2], yv[3]);
    yo[1] = make_float4(yv[4], yv[5], yv[6], yv[7]);
    logdet[i] = logf(prod);
  }
}

extern "C" void kernel_launch(void* const* d_in, const int* in_sizes, int n_in,
                              void* d_out, int out_size, void* d_ws, size_t ws_size,
                              hipStream_t stream) {
  const float* x = (const float*)d_in[0];   // (N_POINTS, 8) f32
  const float* p = (const float*)d_in[1];   // (63, 8) f32
  float* ws = (float*)d_ws;
  const int npts = in_sizes[0] / DIM;
  float* yout   = (float*)d_out;                       // (npts, 8)
  float* ldout  = yout + (size_t)npts * DIM;           // (npts,)

  cdfq_setup_kernel<<<1, 32, 0, stream>>>(p, ws);
  int blocks = (npts + 255) / 256;
  if (blocks < 1) blocks = 1;
  cdfq_main_kernel<<<blocks, 256, 0, stream>>>(x, ws, yout, ldout, npts);
}